// QuantumConv2d_1563368096331
// MI455X (gfx1250) — compile-verified
//
#include <hip/hip_runtime.h>

// QuantumConv2d for MI455X (gfx1250, wave32).
// A = permuted U rows (16x16, constant), B = states of 16 patches,
// D = complex matmul via 4 chained V_WMMA_F32_16X16X4_F32 per real part.

typedef __attribute__((ext_vector_type(2))) float v2f;
typedef __attribute__((ext_vector_type(8))) float v8f;

// CNOT ring permutation: out[i] = in[perm(i)]; composition f12∘f23∘f34∘f41.
__device__ __forceinline__ int perm4(int i) {
  i ^= ((i >> 3) & 1) << 0;  // CNOT(4,1)
  i ^= ((i >> 2) & 1) << 3;  // CNOT(3,4)
  i ^= ((i >> 1) & 1) << 2;  // CNOT(2,3)
  i ^= ((i >> 0) & 1) << 1;  // CNOT(1,2)
  return i;
}

// Build permuted RX-kron matrix: ws[0..255] = Re(U[perm(ch)][k]),
// ws[256..511] = Im(U[perm(ch)][k]); U[i][j] = prod_q rx(w_q)[b_q(i)][b_q(j)],
// rx diag = cos(w/2), offdiag = -i*sin(w/2)  =>  mag * (-i)^popcount(i^j).
__global__ void build_u_kernel(const float* __restrict__ w, float* __restrict__ ws) {
  int t = threadIdx.x;            // 0..255
  int ch = t >> 4, k = t & 15;
  int i = perm4(ch);
  int diff = i ^ k;
  float mag = 1.0f;
#pragma unroll
  for (int q = 0; q < 4; ++q) {
    float hw = 0.5f * w[q];
    mag *= ((diff >> q) & 1) ? __sinf(hw) : __cosf(hw);
  }
  int nf = __popc(diff) & 3;      // (-i)^nf
  float re = (nf == 0) ? mag : ((nf == 2) ? -mag : 0.0f);
  float im = (nf == 1) ? -mag : ((nf == 3) ? mag : 0.0f);
  ws[t] = re;
  ws[256 + t] = im;
}

// 256 threads = 8 waves per block; each wave handles 16 consecutive patches.
// P = 64*128*128 = 1048576 patches -> 65536 tiles -> 8192 blocks.
__global__ __launch_bounds__(256) void qconv_kernel(const float* __restrict__ x,
                                                    const float* __restrict__ ws,
                                                    float* __restrict__ out) {
  const int lane = threadIdx.x & 31;
  const int tile = blockIdx.x * 8 + (threadIdx.x >> 5);
  const int m = lane & 15;        // patch within tile == D column N
  const int h = lane >> 4;        // lane half

  // ---- A fragments: rows of permuted U (M = output channel), constant ----
  // A 16x4 f32 layout: lane L: M = L%16, VGPR0 = K = 4s+2h, VGPR1 = K+1.
  v2f ar[4], ai[4], ain[4];
#pragma unroll
  for (int s = 0; s < 4; ++s) {
    int idx = m * 16 + 4 * s + 2 * h;
    ar[s] = (v2f){ws[idx], ws[idx + 1]};
    float i0 = ws[256 + idx], i1 = ws[256 + idx + 1];
    ai[s]  = (v2f){i0, i1};
    ain[s] = (v2f){-i0, -i1};     // f32 WMMA has no A-neg; pre-negate
  }

  // ---- per-lane patch: decompose p -> (b, ho, wo); load 2x2 patch ----
  const int p   = tile * 16 + m;
  const int b   = p >> 14;        // / (128*128)
  const int rem = p & 16383;
  const int ho  = rem >> 7;
  const int wo  = rem & 127;
  const float* xb = x + ((size_t)b << 16) + (size_t)(ho << 1) * 256 + (wo << 1);
  const float x0 = xb[0], x1 = xb[1], x2 = xb[256], x3 = xb[257];
  const float Q = x0 * x0 + x1 * x1 + x2 * x2 + x3 * x3;

  // ---- B fragments: states for this lane's patch at its 8 K-dims ----
  // B 4x16 f32 layout: lane L: N = L%16, VGPR0 = row K = 4s+2h, VGPR1 = K+1.
  // phase(d) = 0.5*t - 0.25*t^2 + 0.25*Q, t = sum_q sign_q(d)*x_q.
  v2f br[4], bi[4];
#pragma unroll
  for (int s = 0; s < 4; ++s) {
    float sr2[2], si2[2];
#pragma unroll
    for (int j = 0; j < 2; ++j) {
      int d = 4 * s + 2 * h + j;
      float t = (((d >> 0) & 1) ? x0 : -x0)
              + (((d >> 1) & 1) ? x1 : -x1)
              + (((d >> 2) & 1) ? x2 : -x2)
              + (((d >> 3) & 1) ? x3 : -x3);
      float phase = 0.5f * t - 0.25f * t * t + 0.25f * Q;
      float sn, cs;
      __sincosf(phase, &sn, &cs);
      sr2[j] = 0.25f * cs;        // DIM^-0.5 = 1/4
      si2[j] = 0.25f * sn;
    }
    br[s] = (v2f){sr2[0], sr2[1]};
    bi[s] = (v2f){si2[0], si2[1]};
  }

  // ---- complex GEMM: D[ch][p] = sum_k U'[ch][k] * state[p][k] ----
  // Re = Ur*Sr - Ui*Si ; Im = Ur*Si + Ui*Sr ; K=16 as 4 steps of K=4.
  v8f cr = {};
  v8f ci = {};
#pragma unroll
  for (int s = 0; s < 4; ++s) {
    cr = __builtin_amdgcn_wmma_f32_16x16x4_f32(false, ar[s],  false, br[s], (short)0, cr, false, false);
    ci = __builtin_amdgcn_wmma_f32_16x16x4_f32(false, ar[s],  false, bi[s], (short)0, ci, false, false);
    cr = __builtin_amdgcn_wmma_f32_16x16x4_f32(false, ain[s], false, bi[s], (short)0, cr, false, false);
    ci = __builtin_amdgcn_wmma_f32_16x16x4_f32(false, ai[s],  false, br[s], (short)0, ci, false, false);
  }

  // ---- |.|^2 and store; D VGPR r: lanes0-15 ch=r, lanes16-31 ch=r+8,
  // column N = m = wo offset -> contiguous 64B runs per half-wave ----
  const size_t obase = ((size_t)b << 18) + ((size_t)ho << 7) + (size_t)wo;
#pragma unroll
  for (int r = 0; r < 8; ++r) {
    int ch = r + 8 * h;
    float pr = cr[r] * cr[r] + ci[r] * ci[r];
    out[obase + ((size_t)ch << 14)] = pr;
  }
}

extern "C" void kernel_launch(void* const* d_in, const int* in_sizes, int n_in,
                              void* d_out, int out_size, void* d_ws, size_t ws_size,
                              hipStream_t stream) {
  const float* x = (const float*)d_in[0];   // [64,256,256] fp32
  const float* w = (const float*)d_in[1];   // [2,2] fp32
  float* out = (float*)d_out;               // [64,16,128,128] fp32
  float* wsf = (float*)d_ws;                // 512 floats used

  build_u_kernel<<<1, 256, 0, stream>>>(w, wsf);
  // 1048576 patches / 16 per wave / 8 waves per block = 8192 blocks
  qconv_kernel<<<8192, 256, 0, stream>>>(x, wsf, out);
}